// GCN_4604204942041
// MI455X (gfx1250) — compile-verified
//
#include <hip/hip_runtime.h>

// ---------------- problem constants (match reference) ----------------
#define N_NODES 100000
#define N_EDGES 1600000
#define N_GRAPH 64
#define DIM     128
#define HID     128
#define NOUT    10

typedef float        v2f   __attribute__((ext_vector_type(2)));
typedef float        v8f   __attribute__((ext_vector_type(8)));
typedef unsigned int u32x4 __attribute__((ext_vector_type(4)));
typedef int          i32x4 __attribute__((ext_vector_type(4)));
typedef int          i32x8 __attribute__((ext_vector_type(8)));

// -------- Tensor Data Mover availability / arity detection --------
#if defined(__has_builtin)
#if __has_builtin(__builtin_amdgcn_tensor_load_to_lds)
#define GCN_HAS_TDM 1
#endif
#endif
#if defined(GCN_HAS_TDM) && defined(__has_include)
#if __has_include(<hip/amd_detail/amd_gfx1250_TDM.h>)
#define GCN_TDM_6ARG 1   // amdgpu-toolchain (clang-23, therock headers): 6-arg builtin
#endif
#endif

__device__ __forceinline__ void atomic_add_f32(float* p, float v) {
    (void)__hip_atomic_fetch_add(p, v, __ATOMIC_RELAXED, __HIP_MEMORY_SCOPE_AGENT);
}

// ---------------- init: deg=1 (self loop), zero pooling buffers ----------------
__global__ __launch_bounds__(256) void k_init(float* __restrict__ deg,
                                              float* __restrict__ sums,
                                              float* __restrict__ cnt) {
    int i = blockIdx.x * 256 + threadIdx.x;
    if (i < N_NODES)         deg[i]  = 1.0f;     // self-loop contribution
    if (i < N_GRAPH * HID)   sums[i] = 0.0f;
    if (i < N_GRAPH)         cnt[i]  = 0.0f;
}

// ---------------- degree histogram over dst ----------------
__global__ __launch_bounds__(256) void k_degree(const int* __restrict__ ei,
                                                float* __restrict__ deg) {
    int e = blockIdx.x * 256 + threadIdx.x;
    if (e < N_EDGES) atomic_add_f32(&deg[ei[N_EDGES + e]], 1.0f);
}

// ---------------- dis = rsqrt(deg), in place ----------------
__global__ __launch_bounds__(256) void k_rsqrt(float* __restrict__ d) {
    int i = blockIdx.x * 256 + threadIdx.x;
    if (i < N_NODES) d[i] = __frsqrt_rn(d[i]);   // deg >= 1 always
}

// ---------------- GEMM: C[N,128] = A[N,128] @ W[128,128] via fp32 WMMA ----------------
// block = 256 threads = 8 waves; each wave owns one 16-row tile, all 8 col tiles.
// W is staged in LDS (64 KB) once per block via the Tensor Data Mover.
__global__ __launch_bounds__(256) void k_gemm_wmma(const float* __restrict__ Ain,
                                                   const float* __restrict__ Wg,
                                                   float* __restrict__ Cout,
                                                   int nTiles) {
    __shared__ float Wlds[128 * 128];            // 64 KB

#if defined(GCN_HAS_TDM)
    if (threadIdx.x < 32) {                      // wave 0 issues the DMA (EXEC ignored by TDM)
        unsigned long long ga = (unsigned long long)(const void*)Wg;
        unsigned int ldsOff   = (unsigned int)(unsigned long long)(void*)&Wlds[0];
        // D# group0: count=1 | lds_addr | global_addr[56:0] | type=2
        u32x4 g0 = { 1u, ldsOff, (unsigned int)ga,
                     ((unsigned int)(ga >> 32) & 0x01FFFFFFu) | 0x80000000u };
        // D# group1: data_size=4B(code2); tensor 128x128, tile 128x128, stride0=128
        i32x8 g1 = { (int)(2u << 16),        // [17:16] data_size
                     (int)(128u << 16),      // tensor_dim0 low16 @ [63:48]
                     (int)(128u << 16),      // tensor_dim0 hi16(0) | tensor_dim1 low16
                     (int)(128u << 16),      // tensor_dim1 hi16(0) | tile_dim0=128
                     (int)128,               // tile_dim1=128 | tile_dim2=0
                     (int)128,               // tensor_dim0_stride low32
                     0, 0 };
        i32x4 gz = { 0, 0, 0, 0 };           // groups 2/3 unused for 2D tile
#if defined(GCN_TDM_6ARG)
        i32x8 gz8 = { 0, 0, 0, 0, 0, 0, 0, 0 };
        __builtin_amdgcn_tensor_load_to_lds(g0, g1, gz, gz, gz8, 0);
#else
        __builtin_amdgcn_tensor_load_to_lds(g0, g1, gz, gz, 0);
#endif
        __builtin_amdgcn_s_wait_tensorcnt(0);
    }
#else
    for (int i = threadIdx.x; i < (128 * 128) / 4; i += 256)
        ((float4*)Wlds)[i] = ((const float4*)Wg)[i];
#endif
    __syncthreads();

    const int wave  = threadIdx.x >> 5;
    const int lane  = threadIdx.x & 31;
    const int tile  = blockIdx.x * 8 + wave;
    if (tile >= nTiles) return;

    const int m     = lane & 15;                 // row within tile (A) / col within tile (B)
    const int khalf = (lane >> 4) << 1;          // lanes 16-31 hold K = k0+2, k0+3

    const v2f* arow2 = (const v2f*)(Ain + (size_t)(tile * 16 + m) * 128);

    v8f acc[8] = {};
    for (int k0 = 0; k0 < 128; k0 += 4) {
        v2f a = arow2[(k0 + khalf) >> 1];        // A[M=m][K=k0+khalf .. +1]
#pragma unroll
        for (int j = 0; j < 8; ++j) {
            v2f b;
            b.x = Wlds[(k0 + khalf) * 128 + j * 16 + m];       // B[K][N=j*16+m]
            b.y = Wlds[(k0 + khalf + 1) * 128 + j * 16 + m];
            acc[j] = __builtin_amdgcn_wmma_f32_16x16x4_f32(
                false, a, false, b, (short)0, acc[j], false, false);
        }
    }

    // D layout: VGPR r -> lanes 0-15: (M=r, N=lane) ; lanes 16-31: (M=r+8, N=lane-16)
    const int rofs = (lane >> 4) << 3;
#pragma unroll
    for (int j = 0; j < 8; ++j) {
        const int col = j * 16 + m;
#pragma unroll
        for (int r = 0; r < 8; ++r) {
            const int row = tile * 16 + r + rofs;
            Cout[(size_t)row * 128 + col] = acc[j][r];
        }
    }
}

// ---------------- self-loop init: hout[i] = hw[i] * dis[i]^2 ----------------
__global__ __launch_bounds__(256) void k_selfloop(const float* __restrict__ hw,
                                                  const float* __restrict__ dis,
                                                  float* __restrict__ hout) {
    size_t idx = (size_t)blockIdx.x * 256 + threadIdx.x;   // N*128 elements
    int i = (int)(idx >> 7);
    float s = dis[i];
    hout[idx] = hw[idx] * s * s;
}

// ---------------- edge scatter: hout[dst] += hw[src] * dis[src]*dis[dst] ----------------
// one edge per wave; 4 features per lane (wave32 * 4 = 128)
__global__ __launch_bounds__(256) void k_edge(const int* __restrict__ ei,
                                              const float* __restrict__ dis,
                                              const float* __restrict__ hw,
                                              float* __restrict__ hout) {
    int e    = (int)(((size_t)blockIdx.x * 256 + threadIdx.x) >> 5);
    int lane = threadIdx.x & 31;
    if (e >= N_EDGES) return;
    int s = ei[e];
    int d = ei[N_EDGES + e];
    float w = dis[s] * dis[d];
    float4 v = ((const float4*)(hw + (size_t)s * 128))[lane];
    float* o = hout + (size_t)d * 128 + lane * 4;
    atomic_add_f32(o + 0, v.x * w);
    atomic_add_f32(o + 1, v.y * w);
    atomic_add_f32(o + 2, v.z * w);
    atomic_add_f32(o + 3, v.w * w);
}

// ---------------- bias (+ optional ReLU) ----------------
__global__ __launch_bounds__(256) void k_bias_act(float* __restrict__ h,
                                                  const float* __restrict__ b,
                                                  int doRelu) {
    size_t idx = (size_t)blockIdx.x * 256 + threadIdx.x;   // N*128 elements
    int c = (int)(idx & 127);
    float v = h[idx] + b[c];
    h[idx] = doRelu ? fmaxf(v, 0.0f) : v;
}

// ---------------- global mean pool (accumulate) ----------------
__global__ __launch_bounds__(256) void k_pool(const float* __restrict__ h,
                                              const int* __restrict__ batch,
                                              float* __restrict__ sums,
                                              float* __restrict__ cnt) {
    int i    = (int)(((size_t)blockIdx.x * 256 + threadIdx.x) >> 5);  // node
    int lane = threadIdx.x & 31;
    if (i >= N_NODES) return;
    int g = batch[i];
    float4 v = ((const float4*)(h + (size_t)i * 128))[lane];
    float* s = sums + (size_t)g * 128 + lane * 4;
    atomic_add_f32(s + 0, v.x);
    atomic_add_f32(s + 1, v.y);
    atomic_add_f32(s + 2, v.z);
    atomic_add_f32(s + 3, v.w);
    if (lane == 0) atomic_add_f32(&cnt[g], 1.0f);
}

// ---------------- head: out[g,o] = (sums[g]/cnt[g]) @ lin_W + lin_b ----------------
__global__ __launch_bounds__(256) void k_head(const float* __restrict__ sums,
                                              const float* __restrict__ cnt,
                                              const float* __restrict__ linW,
                                              const float* __restrict__ linb,
                                              float* __restrict__ out) {
    int t = blockIdx.x * 256 + threadIdx.x;
    if (t >= N_GRAPH * NOUT) return;
    int g = t / NOUT, o = t % NOUT;
    float inv = 1.0f / fmaxf(cnt[g], 1.0f);
    float acc = 0.0f;
    const float* sg = sums + (size_t)g * 128;
#pragma unroll 8
    for (int k = 0; k < 128; ++k) acc += sg[k] * linW[k * NOUT + o];
    out[t] = acc * inv + linb[o];
}

// ================================================================
extern "C" void kernel_launch(void* const* d_in, const int* in_sizes, int n_in,
                              void* d_out, int out_size, void* d_ws, size_t ws_size,
                              hipStream_t stream) {
    (void)in_sizes; (void)n_in; (void)out_size; (void)ws_size;

    const float* x     = (const float*)d_in[0];
    const int*   ei    = (const int*)  d_in[1];
    const int*   batch = (const int*)  d_in[2];
    const float* W1    = (const float*)d_in[3];
    const float* b1    = (const float*)d_in[4];
    const float* W2    = (const float*)d_in[5];
    const float* b2    = (const float*)d_in[6];
    const float* W3    = (const float*)d_in[7];
    const float* b3    = (const float*)d_in[8];
    const float* linW  = (const float*)d_in[9];
    const float* linb  = (const float*)d_in[10];
    float*       out   = (float*)d_out;

    // workspace layout (bytes), all offsets 128B aligned; total ~98 MB
    char* ws = (char*)d_ws;
    float* dis  = (float*)(ws + 0);                       // N floats (deg -> rsqrt)
    float* hw   = (float*)(ws + 400384);                  // N*128 floats
    float* hbuf = (float*)(ws + 400384 + 51200000);       // N*128 floats
    float* sums = (float*)(ws + 400384 + 2 * 51200000ULL);// G*128 floats
    float* cnt  = (float*)(ws + 400384 + 2 * 51200000ULL + 32768); // G floats

    const int nTiles     = N_NODES / 16;                  // 6250
    const int gemmBlocks = (nTiles + 7) / 8;              // 782
    const int elemBlocks = (N_NODES * 128) / 256;         // 50000
    const int edgeBlocks = (N_EDGES * 32) / 256;          // 200000
    const int poolBlocks = (N_NODES * 32) / 256;          // 12500
    const int nBlocks    = (N_NODES + 255) / 256;         // 391

    // --- normalization table ---
    k_init  <<<nBlocks, 256, 0, stream>>>(dis, sums, cnt);
    k_degree<<<(N_EDGES + 255) / 256, 256, 0, stream>>>(ei, dis);
    k_rsqrt <<<nBlocks, 256, 0, stream>>>(dis);

    // --- layer 1 ---
    k_gemm_wmma<<<gemmBlocks, 256, 0, stream>>>(x, W1, hw, nTiles);
    k_selfloop <<<elemBlocks, 256, 0, stream>>>(hw, dis, hbuf);
    k_edge     <<<edgeBlocks, 256, 0, stream>>>(ei, dis, hw, hbuf);
    k_bias_act <<<elemBlocks, 256, 0, stream>>>(hbuf, b1, 1);

    // --- layer 2 ---
    k_gemm_wmma<<<gemmBlocks, 256, 0, stream>>>(hbuf, W2, hw, nTiles);
    k_selfloop <<<elemBlocks, 256, 0, stream>>>(hw, dis, hbuf);
    k_edge     <<<edgeBlocks, 256, 0, stream>>>(ei, dis, hw, hbuf);
    k_bias_act <<<elemBlocks, 256, 0, stream>>>(hbuf, b2, 1);

    // --- layer 3 (no relu) ---
    k_gemm_wmma<<<gemmBlocks, 256, 0, stream>>>(hbuf, W3, hw, nTiles);
    k_selfloop <<<elemBlocks, 256, 0, stream>>>(hw, dis, hbuf);
    k_edge     <<<edgeBlocks, 256, 0, stream>>>(ei, dis, hw, hbuf);
    k_bias_act <<<elemBlocks, 256, 0, stream>>>(hbuf, b3, 0);

    // --- pool + head ---
    k_pool<<<poolBlocks, 256, 0, stream>>>(hbuf, batch, sums, cnt);
    k_head<<<(N_GRAPH * NOUT + 255) / 256, 256, 0, stream>>>(sums, cnt, linW, linb, out);
}